// ARModel_88880053223563
// MI455X (gfx1250) — compile-verified
//
#include <hip/hip_runtime.h>

// out[b,t,n,0] = x[b,t-1,n,0] * w + bias  (t >= 1);  out[b,0,n,0] = 0
// Flat:  out[i] = (t==0) ? 0 : fma(x[i-N], w, bias),  t = (i / N) % T
//
// Memory-bound streaming kernel: 147.5 MB in + 147.5 MB out.
// b128 loads/stores; NT hint on the store so the write-once output does not
// evict the (L2-resident-capable, 147.5MB < 192MB) input from global L2.

#define AR_N 2000
#define AR_T 288
#define AR_N4 (AR_N / 4)   // 500 vec4 per (b,t) row — divisible, so a vec4
                           // never straddles a t boundary and stays 16B aligned.

// Native clang vector type: valid operand for __builtin_nontemporal_* and
// lowers to global_{load,store}_b128.
typedef float v4f __attribute__((ext_vector_type(4)));

__global__ __launch_bounds__(256)
void ar_shift_scale_f32(const float* __restrict__ x,
                        const float* __restrict__ w,
                        const float* __restrict__ bias,
                        float* __restrict__ out,
                        int total4)
{
    const float wv = w[0];
    const float bv = bias[0];

    const v4f* __restrict__ xp = reinterpret_cast<const v4f*>(x);
    v4f*       __restrict__ op = reinterpret_cast<v4f*>(out);

    const int stride = (int)(gridDim.x * blockDim.x);
    for (int i4 = (int)(blockIdx.x * blockDim.x + threadIdx.x);
         i4 < total4; i4 += stride)
    {
        const unsigned idx = (unsigned)i4 * 4u;
        const unsigned t   = (idx / (unsigned)AR_N) % (unsigned)AR_T;

        v4f r;
        if (t != 0u) {
            // Regular-temporal load: keep x resident in the 192MB L2.
            const v4f v = xp[i4 - AR_N4];
            r.x = fmaf(v.x, wv, bv);
            r.y = fmaf(v.y, wv, bv);
            r.z = fmaf(v.z, wv, bv);
            r.w = fmaf(v.w, wv, bv);
        } else {
            r = (v4f)(0.f);
        }
        // Non-temporal store: output is write-once, never re-read; don't let
        // it thrash L2 (in+out = 295MB > 192MB, input alone fits).
        __builtin_nontemporal_store(r, op + i4);
    }
}

// Scalar tail guard (only used if out_size % 4 != 0; here 36,864,000 % 4 == 0,
// so this launches with zero blocks in practice — kept for robustness).
__global__ void ar_shift_scale_tail(const float* __restrict__ x,
                                    const float* __restrict__ w,
                                    const float* __restrict__ bias,
                                    float* __restrict__ out,
                                    int start, int total)
{
    int i = start + (int)(blockIdx.x * blockDim.x + threadIdx.x);
    if (i >= total) return;
    const unsigned t = ((unsigned)i / (unsigned)AR_N) % (unsigned)AR_T;
    out[i] = (t == 0u) ? 0.f : fmaf(x[i - AR_N], w[0], bias[0]);
}

extern "C" void kernel_launch(void* const* d_in, const int* in_sizes, int n_in,
                              void* d_out, int out_size, void* d_ws, size_t ws_size,
                              hipStream_t stream) {
    (void)in_sizes; (void)n_in; (void)d_ws; (void)ws_size;

    const float* x    = (const float*)d_in[0];  // (64,288,2000,1) f32
    const float* w    = (const float*)d_in[1];  // (1,1,1) f32
    const float* bias = (const float*)d_in[2];  // (1,)   f32
    float*       out  = (float*)d_out;          // (64,288,2000,1) f32

    const int total  = out_size;          // 36,864,000
    const int total4 = total / 4;         // 9,216,000 vec4
    const int block  = 256;               // 8 wave32 per block
    const int grid   = (total4 + block - 1) / block;   // 36,000 blocks

    if (grid > 0) {
        ar_shift_scale_f32<<<grid, block, 0, stream>>>(x, w, bias, out, total4);
    }

    const int tail = total - total4 * 4;
    if (tail > 0) {
        ar_shift_scale_tail<<<(tail + 63) / 64, 64, 0, stream>>>(
            x, w, bias, out, total4 * 4, total);
    }
}